// SpatiotemporalUncertaintyLoss_24790551232751
// MI455X (gfx1250) — compile-verified
//
#include <hip/hip_runtime.h>
#include <cmath>

// Problem sizes (fixed by the reference)
#define N_L_C  131072
#define E_S_C  2097152
#define N_R_C  8192
#define E_T_C  65536
#define E_RL_C 131072
#define N_G_C  8192

#define BIGF   1.0e30f
#define GHOSTF 2.0f

typedef float v2f __attribute__((ext_vector_type(2)));
typedef float v4f __attribute__((ext_vector_type(4)));
typedef float v8f __attribute__((ext_vector_type(8)));

// ---------------------------------------------------------------------------
// wave32 helpers
// ---------------------------------------------------------------------------
__device__ __forceinline__ float wave_sum(float v) {
#pragma unroll
  for (int off = 16; off >= 1; off >>= 1)
    v += __shfl_xor(v, off, 32);
  return v;
}

// ---------------------------------------------------------------------------
// Lidar: scatter pos[src], intensity[src], count into dst
// ---------------------------------------------------------------------------
__global__ __launch_bounds__(256)
void k_lidar_edges(const int* __restrict__ edges, const float* __restrict__ pos,
                   const float* __restrict__ x, float* __restrict__ sum_pos,
                   float* __restrict__ sum_int, float* __restrict__ cnt, int n_e) {
  int e = blockIdx.x * blockDim.x + threadIdx.x;
  if (e >= n_e) return;
  int s = edges[e];
  int d = edges[n_e + e];
  atomicAdd(&sum_pos[d * 3 + 0], pos[s * 3 + 0]);
  atomicAdd(&sum_pos[d * 3 + 1], pos[s * 3 + 1]);
  atomicAdd(&sum_pos[d * 3 + 2], pos[s * 3 + 2]);
  atomicAdd(&sum_int[d], x[s * 4 + 2]);
  atomicAdd(&cnt[d], 1.0f);
}

__global__ __launch_bounds__(256)
void k_lidar_nodes(const float* __restrict__ pos, const float* __restrict__ x,
                   const float* __restrict__ logv, const float* __restrict__ sum_pos,
                   const float* __restrict__ sum_int, const float* __restrict__ cnt,
                   float lmin, float lmax, float* __restrict__ acc, int n) {
  int i = blockIdx.x * blockDim.x + threadIdx.x;
  float val = 0.0f;
  if (i < n) {
    float lv = fminf(fmaxf(logv[i], lmin), lmax);
    float c = fmaxf(cnt[i], 1.0f);
    float mx = sum_pos[i * 3 + 0] / c;
    float my = sum_pos[i * 3 + 1] / c;
    float mz = sum_pos[i * 3 + 2] / c;
    float mi = sum_int[i] / c;
    float dx = pos[i * 3 + 0] - mx, dy = pos[i * 3 + 1] - my, dz = pos[i * 3 + 2] - mz;
    float di = x[i * 4 + 2] - mi;
    float comb = dx * dx + dy * dy + dz * dz + di * di;  // W_L_INT = 1
    val = 0.5f * expf(-lv) * comb + 0.5f * lv;
  }
  float s = wave_sum(val);
  if ((threadIdx.x & 31) == 0) atomicAdd(acc, s);
}

// ---------------------------------------------------------------------------
// Pack gt for the WMMA loop: gpack[j] = (-2gx, -2gy, -2gz, |g|^2)
// ---------------------------------------------------------------------------
__global__ __launch_bounds__(256)
void k_gpack(const float* __restrict__ gt, v4f* __restrict__ gpack, int n) {
  int i = blockIdx.x * blockDim.x + threadIdx.x;
  if (i >= n) return;
  float gx = gt[i * 3 + 0], gy = gt[i * 3 + 1], gz = gt[i * 3 + 2];
  v4f p;
  p.x = -2.0f * gx;
  p.y = -2.0f * gy;
  p.z = -2.0f * gz;
  p.w = gx * gx + gy * gy + gz * gz;
  gpack[i] = p;
}

// ---------------------------------------------------------------------------
// Radar temporal edges: unit motion vectors scattered into dst
// ---------------------------------------------------------------------------
__global__ __launch_bounds__(256)
void k_radar_temporal(const int* __restrict__ edges, const float* __restrict__ pos,
                      float* __restrict__ dsum, float* __restrict__ dcnt, int n_e) {
  int e = blockIdx.x * blockDim.x + threadIdx.x;
  if (e >= n_e) return;
  int s = edges[e];
  int d = edges[n_e + e];
  float mx = pos[d * 3 + 0] - pos[s * 3 + 0];
  float my = pos[d * 3 + 1] - pos[s * 3 + 1];
  float mz = pos[d * 3 + 2] - pos[s * 3 + 2];
  float nrm = sqrtf(mx * mx + my * my + mz * mz) + 1e-9f;
  atomicAdd(&dsum[d * 3 + 0], mx / nrm);
  atomicAdd(&dsum[d * 3 + 1], my / nrm);
  atomicAdd(&dsum[d * 3 + 2], mz / nrm);
  atomicAdd(&dcnt[d], 1.0f);
}

// ---------------------------------------------------------------------------
// Radar per-node: dvec, pos_phy, |pos_phy|^2, denom, reg sum
// ---------------------------------------------------------------------------
__global__ __launch_bounds__(256)
void k_radar_nodes(const float* __restrict__ pos, const float* __restrict__ x,
                   const float* __restrict__ logv, const float* __restrict__ dsum,
                   const float* __restrict__ dcnt, float rmin, float rmax,
                   float* __restrict__ pos_phy, float* __restrict__ p2,
                   float* __restrict__ denomv, float* __restrict__ acc_reg, int n) {
  int i = blockIdx.x * blockDim.x + threadIdx.x;
  float reg = 0.0f;
  if (i < n) {
    float c = fmaxf(dcnt[i], 1.0f);
    float dx = dsum[i * 3 + 0] / c, dy = dsum[i * 3 + 1] / c, dz = dsum[i * 3 + 2] / c;
    float nrm = sqrtf(dx * dx + dy * dy + dz * dz) + 1e-9f;
    dx /= nrm; dy /= nrm; dz /= nrm;
    float spd = fabsf(x[i * 4 + 2]);
    float px = pos[i * 3 + 0] + spd * dx * 0.1f;
    float py = pos[i * 3 + 1] + spd * dy * 0.1f;
    float pz = pos[i * 3 + 2] + spd * dz * 0.1f;
    pos_phy[i * 3 + 0] = px; pos_phy[i * 3 + 1] = py; pos_phy[i * 3 + 2] = pz;
    p2[i] = px * px + py * py + pz * pz;
    float lv = fminf(fmaxf(logv[i], rmin), rmax);
    denomv[i] = 2.0f * expf(lv) * 0.010000000707805157f + 1e-9f;  // SAFE_DT^2
    reg = 0.5f * lv;
  }
  float s = wave_sum(reg);
  if ((threadIdx.x & 31) == 0) atomicAdd(acc_reg, s);
}

// ---------------------------------------------------------------------------
// Radar spatial edges: radar->lidar squared distance scattered into src
// ---------------------------------------------------------------------------
__global__ __launch_bounds__(256)
void k_radar_spatial(const int* __restrict__ src_r, const int* __restrict__ dst_l,
                     const float* __restrict__ rpos, const float* __restrict__ lpos,
                     float* __restrict__ ssum, float* __restrict__ scnt, int n_e) {
  int e = blockIdx.x * blockDim.x + threadIdx.x;
  if (e >= n_e) return;
  int s = src_r[e];
  int d = dst_l[e];
  float dx = rpos[s * 3 + 0] - lpos[d * 3 + 0];
  float dy = rpos[s * 3 + 1] - lpos[d * 3 + 1];
  float dz = rpos[s * 3 + 2] - lpos[d * 3 + 2];
  atomicAdd(&ssum[s], dx * dx + dy * dy + dz * dz);
  atomicAdd(&scnt[s], 1.0f);
}

__global__ __launch_bounds__(256)
void k_radar_spat_nodes(const float* __restrict__ ssum, const float* __restrict__ scnt,
                        const float* __restrict__ denomv, float* __restrict__ acc_spat,
                        int n) {
  int i = blockIdx.x * blockDim.x + threadIdx.x;
  float val = 0.0f;
  if (i < n) {
    float c = scnt[i];
    float cm = fmaxf(c, 1.0f);
    float spat = (c > 0.0f) ? (ssum[i] / (cm * cm)) : GHOSTF;
    val = spat / denomv[i];
  }
  float s = wave_sum(val);
  if ((threadIdx.x & 31) == 0) atomicAdd(acc_spat, s);
}

// ---------------------------------------------------------------------------
// WMMA batched pair-distance + masked row-min.
// One wave per 16-row radar tile; V_WMMA_F32_16X16X4_F32 computes the full
// 16x16 d2 tile per iteration: D = A(pos_phy) x B(gpack.xyz) + C(p2 + gpack.w).
// A 16x4 f32 layout: lanes 0-15 hold (K0,K1), lanes 16-31 hold (K2,K3=0).
// B 4x16 f32 layout mirrors A with lane = column N.
// D layout: VGPR r, lanes 0-15 -> row M=r; lanes 16-31 -> row M=r+8.
// gpack holds -2*gt so D = p2 + g2 - 2*p.g = d2 directly.
// The B operand is one b128 load + register selects: no divergent loads.
// ---------------------------------------------------------------------------
__global__ __launch_bounds__(256)
void k_radar_min_wmma(const float* __restrict__ pos_phy, const float* __restrict__ p2,
                      const int* __restrict__ rbatch, const v4f* __restrict__ gpack,
                      const int* __restrict__ gbatch, const float* __restrict__ denomv,
                      float* __restrict__ acc_temp, int n_r, int n_g) {
  const int wave = threadIdx.x >> 5;
  const int lane = threadIdx.x & 31;
  const int half = lane >> 4;
  const int ln16 = lane & 15;
  const int base = (blockIdx.x * (blockDim.x >> 5) + wave) * 16;
  if (base >= n_r) return;  // wave-uniform

  // A operand: one row of pos_phy per lane%16; K split across lane halves, K=3 padded.
  const int rowA = base + ln16;
  v2f a;
  a.x = pos_phy[rowA * 3 + (half ? 2 : 0)];
  a.y = half ? 0.0f : pos_phy[rowA * 3 + 1];

  float p2row[8];
  int rb[8];
#pragma unroll
  for (int r = 0; r < 8; ++r) {
    int m = base + r + 8 * half;
    p2row[r] = p2[m];
    rb[r] = rbatch[m];
  }

  v8f mn;
#pragma unroll
  for (int r = 0; r < 8; ++r) mn[r] = BIGF;

#pragma unroll 2
  for (int jt = 0; jt < n_g; jt += 16) {
    const int col = jt + ln16;
    const v4f g = gpack[col];          // single b128 load per lane per tile
    const int gb = gbatch[col];
    v2f b;
    b.x = half ? g.z : g.x;            // register selects, no divergent loads
    b.y = half ? 0.0f : g.y;
    v8f c;
#pragma unroll
    for (int r = 0; r < 8; ++r) c[r] = p2row[r] + g.w;
    // D = p2 + g2 - 2*p.g  (exact f32, matching the reference expansion)
    v8f d = __builtin_amdgcn_wmma_f32_16x16x4_f32(false, a, false, b, (short)0, c,
                                                  false, false);
#pragma unroll
    for (int r = 0; r < 8; ++r) {
      float v = fmaxf(d[r], 0.0f);
      v = (rb[r] == gb) ? v : BIGF;
      mn[r] = fminf(mn[r], v);
    }
  }

  // min-reduce each row across its 16 lanes
#pragma unroll
  for (int r = 0; r < 8; ++r) {
#pragma unroll
    for (int off = 8; off >= 1; off >>= 1)
      mn[r] = fminf(mn[r], __shfl_xor(mn[r], off, 16));
  }

  if (ln16 == 0) {  // lanes 0 (rows base..base+7) and 16 (rows base+8..base+15)
    float local = 0.0f;
#pragma unroll
    for (int r = 0; r < 8; ++r) {
      int m = base + r + 8 * half;
      float v = mn[r];
      float err = (v < 0.5f * BIGF) ? v : 0.0f;  // has_gt handling
      local += err / denomv[m];
    }
    atomicAdd(acc_temp, local);
  }
}

// ---------------------------------------------------------------------------
// Final scalar combine
// ---------------------------------------------------------------------------
__global__ void k_finalize(const float* __restrict__ acc, float* __restrict__ out) {
  const float inv_nl = 1.0f / (float)N_L_C;
  const float inv_nr = 1.0f / (float)N_R_C;
  float lid = acc[0] * inv_nl;  // W_LIDAR = 1
  float r1 = (acc[1] * inv_nr) * 1.0f + (acc[2] * inv_nr) * 0.1f + acc[3] * inv_nr;
  float r2 = (acc[4] * inv_nr) * 1.0f + (acc[5] * inv_nr) * 0.1f + acc[6] * inv_nr;
  out[0] = lid + r1 + r2;  // W_RADAR = 1
}

// ---------------------------------------------------------------------------
// Host
// ---------------------------------------------------------------------------
extern "C" void kernel_launch(void* const* d_in, const int* in_sizes, int n_in,
                              void* d_out, int out_size, void* d_ws, size_t ws_size,
                              hipStream_t stream) {
  const float* lidar_pos = (const float*)d_in[0];
  const float* lidar_x = (const float*)d_in[1];
  const float* log_var_lidar = (const float*)d_in[2];
  const float* radar_pos[2] = {(const float*)d_in[3], (const float*)d_in[6]};
  const float* radar_x[2] = {(const float*)d_in[4], (const float*)d_in[7]};
  const float* log_var_radar[2] = {(const float*)d_in[5], (const float*)d_in[8]};
  const float* gt_pos = (const float*)d_in[9];
  const int* edge_spatial = (const int*)d_in[10];
  const int* edge_temporal[2] = {(const int*)d_in[11], (const int*)d_in[12]};
  const int* rl_src[2] = {(const int*)d_in[13], (const int*)d_in[15]};
  const int* rl_dst[2] = {(const int*)d_in[14], (const int*)d_in[16]};
  const int* radar_batch[2] = {(const int*)d_in[17], (const int*)d_in[18]};
  const int* gt_batch = (const int*)d_in[19];
  float* out = (float*)d_out;

  // clip bounds computed in double on host, cast to f32 (matches reference)
  const float l_min = (float)(2.0 * std::log(0.03 / 10.0 + 1e-9));
  const float l_max = (float)(2.0 * std::log(0.2 / 10.0 + 1e-9));
  const float r_min = (float)(2.0 * std::log(0.1 / 5.0 + 1e-9));
  const float r_max = (float)(2.0 * std::log(3.0 / 5.0 + 1e-9));

  // workspace layout (floats); base is 256B-aligned, gpack offset stays 16B-aligned
  float* ws = (float*)d_ws;
  size_t off = 0;
  float* acc = ws + off; off += 16;                 // [0] lidar; per radar k: [1+3k] temp, [2+3k] spat, [3+3k] reg
  float* lsp = ws + off; off += (size_t)N_L_C * 3;  // lidar sum_pos
  float* lsi = ws + off; off += N_L_C;              // lidar sum_int
  float* lcn = ws + off; off += N_L_C;              // lidar count
  v4f* gpack = (v4f*)(ws + off); off += (size_t)N_G_C * 4;
  float* dsum[2]; float* dcnt[2]; float* ssum[2]; float* scnt[2];
  float* pphy[2]; float* p2[2]; float* denomv[2];
  for (int k = 0; k < 2; ++k) {
    dsum[k] = ws + off; off += (size_t)N_R_C * 3;
    dcnt[k] = ws + off; off += N_R_C;
    ssum[k] = ws + off; off += N_R_C;
    scnt[k] = ws + off; off += N_R_C;
    pphy[k] = ws + off; off += (size_t)N_R_C * 3;
    p2[k] = ws + off; off += N_R_C;
    denomv[k] = ws + off; off += N_R_C;
  }
  hipMemsetAsync(d_ws, 0, off * sizeof(float), stream);

  // Lidar term
  k_lidar_edges<<<E_S_C / 256, 256, 0, stream>>>(edge_spatial, lidar_pos, lidar_x,
                                                 lsp, lsi, lcn, E_S_C);
  k_lidar_nodes<<<N_L_C / 256, 256, 0, stream>>>(lidar_pos, lidar_x, log_var_lidar,
                                                 lsp, lsi, lcn, l_min, l_max,
                                                 &acc[0], N_L_C);
  // packed gt operands (shared by both radars)
  k_gpack<<<N_G_C / 256, 256, 0, stream>>>(gt_pos, gpack, N_G_C);

  // Radar terms
  for (int k = 0; k < 2; ++k) {
    k_radar_temporal<<<E_T_C / 256, 256, 0, stream>>>(edge_temporal[k], radar_pos[k],
                                                      dsum[k], dcnt[k], E_T_C);
    k_radar_nodes<<<N_R_C / 256, 256, 0, stream>>>(radar_pos[k], radar_x[k],
                                                   log_var_radar[k], dsum[k], dcnt[k],
                                                   r_min, r_max, pphy[k], p2[k],
                                                   denomv[k], &acc[3 + 3 * k], N_R_C);
    k_radar_spatial<<<E_RL_C / 256, 256, 0, stream>>>(rl_src[k], rl_dst[k],
                                                      radar_pos[k], lidar_pos,
                                                      ssum[k], scnt[k], E_RL_C);
    k_radar_spat_nodes<<<N_R_C / 256, 256, 0, stream>>>(ssum[k], scnt[k], denomv[k],
                                                        &acc[2 + 3 * k], N_R_C);
    // 16 rows per wave, 8 waves per block -> 128 rows per block
    k_radar_min_wmma<<<N_R_C / 128, 256, 0, stream>>>(pphy[k], p2[k], radar_batch[k],
                                                      gpack, gt_batch, denomv[k],
                                                      &acc[1 + 3 * k], N_R_C, N_G_C);
  }

  k_finalize<<<1, 1, 0, stream>>>(acc, out);
}